// NormalShader_72086731096585
// MI455X (gfx1250) — compile-verified
//
#include <hip/hip_runtime.h>
#include <math.h>

// Native ext-vectors so __builtin_nontemporal_load/store accept them.
typedef float f4 __attribute__((ext_vector_type(4)));
typedef int   i4 __attribute__((ext_vector_type(4)));

#define KF 8
#define SIGMA_INV 1.0e4f          // 1/SIGMA
#define GAMMA_INV 1.0e4f          // 1/GAMMA
#define ZFARC 100.0f
#define ZSCALE (1.0f / 99.0f)     // 1/(ZFAR-ZNEAR)
#define EPSV 1e-10f

// ---------------------------------------------------------------------------
// Kernel A: per face-vertex, gather vertex color, remap + renormalize, store
// into the face-color table in d_ws. STRIDE=4 pads each vertex color to 16B so
// the shade kernel can gather with one global_load_b128 per vertex.
// ---------------------------------------------------------------------------
template <int STRIDE>
__global__ __launch_bounds__(256) void build_face_tex_k(
    const float* __restrict__ vtex,   // (V,3)
    const int*   __restrict__ faces,  // (F,3) flattened -> F*3 entries
    float*       __restrict__ ft,     // (F*3, STRIDE)
    int F3)
{
  int t = blockIdx.x * blockDim.x + threadIdx.x;
  if (t >= F3) return;
  int vi = faces[t];
  float x = vtex[3 * (size_t)vi + 0];
  float y = vtex[3 * (size_t)vi + 1];
  float z = vtex[3 * (size_t)vi + 2];
  x = (x + 1.0f) * 0.5f;
  y = (y + 1.0f) * 0.5f;
  z = (z + 3.0f) * 0.25f;
  // ||c||^2 is bounded far from denormals (z-term alone ~[0.06,1.6]):
  // raw v_rsq_f32, no denorm-rescale guard needed.
  float inv = __builtin_amdgcn_rsqf(fmaf(x, x, fmaf(y, y, z * z)));
  x *= inv; y *= inv; z *= inv;
  if (STRIDE == 4) {
    f4 c = {x, y, z, 0.0f};
    *((f4*)ft + (size_t)t) = c;   // regular (RT) store: table stays in L2
  } else {
    ft[3 * (size_t)t + 0] = x;
    ft[3 * (size_t)t + 1] = y;
    ft[3 * (size_t)t + 2] = z;
  }
}

// ---------------------------------------------------------------------------
// Kernel B: one thread per pixel. Streams (NT hint) per-pixel inputs, gathers
// face colors from the L2-resident table (RT), softmax-blends over K=8.
// Fully branchless: masked slots gather face 0 (like the reference) and
// contribute with weight 0, so all 24 b128 gathers are unconditional and can
// be clustered/hoisted by the scheduler to hide latency.
// ---------------------------------------------------------------------------
template <int STRIDE>
__global__ __launch_bounds__(256) void shade_k(
    const int*   __restrict__ p2f,    // (NHW, 8)
    const float* __restrict__ bary,   // (NHW, 8, 3)
    const float* __restrict__ dists,  // (NHW, 8)
    const float* __restrict__ zbuf,   // (NHW, 8)
    const float* __restrict__ ft,     // face color table
    float*       __restrict__ out,    // (NHW, 4)
    int NHW)
{
  int p = blockIdx.x * blockDim.x + threadIdx.x;
  if (p >= NHW) return;

  // --- streamed, read-once inputs: non-temporal b128 loads -----------------
  const i4* fp = (const i4*)p2f + (size_t)p * 2;
  i4 fa = __builtin_nontemporal_load(fp);
  i4 fb = __builtin_nontemporal_load(fp + 1);
  int fk[KF] = {fa.x, fa.y, fa.z, fa.w, fb.x, fb.y, fb.z, fb.w};

  const f4* dp = (const f4*)dists + (size_t)p * 2;
  f4 d0 = __builtin_nontemporal_load(dp);
  f4 d1 = __builtin_nontemporal_load(dp + 1);
  float dk[KF] = {d0.x, d0.y, d0.z, d0.w, d1.x, d1.y, d1.z, d1.w};

  const f4* zp = (const f4*)zbuf + (size_t)p * 2;
  f4 z0 = __builtin_nontemporal_load(zp);
  f4 z1 = __builtin_nontemporal_load(zp + 1);
  float zk[KF] = {z0.x, z0.y, z0.z, z0.w, z1.x, z1.y, z1.z, z1.w};

  const f4* bp = (const f4*)bary + (size_t)p * 6;
  f4 b0 = __builtin_nontemporal_load(bp + 0);
  f4 b1 = __builtin_nontemporal_load(bp + 1);
  f4 b2 = __builtin_nontemporal_load(bp + 2);
  f4 b3 = __builtin_nontemporal_load(bp + 3);
  f4 b4 = __builtin_nontemporal_load(bp + 4);
  f4 b5 = __builtin_nontemporal_load(bp + 5);
  float bk[24] = {b0.x, b0.y, b0.z, b0.w, b1.x, b1.y, b1.z, b1.w,
                  b2.x, b2.y, b2.z, b2.w, b3.x, b3.y, b3.z, b3.w,
                  b4.x, b4.y, b4.z, b4.w, b5.x, b5.y, b5.z, b5.w};

  // --- unconditional gathers, clamped index (reference: idx=where(mask,f,0))
  float cx[KF][3], cy[KF][3], cz[KF][3];
#pragma unroll
  for (int k = 0; k < KF; ++k) {
    int idx = fk[k] >= 0 ? fk[k] : 0;          // v_max_i32
    if (STRIDE == 4) {
      const f4* c = (const f4*)ft + (size_t)idx * 3;
      f4 c0 = c[0], c1 = c[1], c2 = c[2];      // 3x global_load_b128
      cx[k][0] = c0.x; cy[k][0] = c0.y; cz[k][0] = c0.z;
      cx[k][1] = c1.x; cy[k][1] = c1.y; cz[k][1] = c1.z;
      cx[k][2] = c2.x; cy[k][2] = c2.y; cz[k][2] = c2.z;
    } else {
      const float* c = ft + (size_t)idx * 9;
      cx[k][0] = c[0]; cy[k][0] = c[1]; cz[k][0] = c[2];
      cx[k][1] = c[3]; cy[k][1] = c[4]; cz[k][1] = c[5];
      cx[k][2] = c[6]; cy[k][2] = c[7]; cz[k][2] = c[8];
    }
  }

  // --- pass 1: prob, z_inv, running max, alpha (overlaps gather latency) ---
  float prob[KF], zinv[KF];
  float zmax = EPSV;     // z_inv_max = max(max_k z_inv, EPS)
  float alpha = 1.0f;
#pragma unroll
  for (int k = 0; k < KF; ++k) {
    bool m = fk[k] >= 0;
    float e  = __expf(dk[k] * SIGMA_INV);                    // sigmoid(-d/s)
    float pr = m ? __builtin_amdgcn_rcpf(1.0f + e) : 0.0f;
    prob[k] = pr;
    alpha *= (1.0f - pr);
    float zi = m ? (ZFARC - zk[k]) * ZSCALE : 0.0f;
    zinv[k] = zi;
    zmax = fmaxf(zmax, zi);
  }

  float delta = __expf((EPSV - zmax) * GAMMA_INV);
  float denom = delta;
  float r = 0.0f, g = 0.0f, b = 0.0f;

  // --- pass 2: blend (branchless; prob[k]==0 zeroes masked contributions) --
#pragma unroll
  for (int k = 0; k < KF; ++k) {
    float w = prob[k] * __expf((zinv[k] - zmax) * GAMMA_INV);
    float w0 = bk[3 * k + 0], w1 = bk[3 * k + 1], w2 = bk[3 * k + 2];
    float tx = fmaf(w0, cx[k][0], fmaf(w1, cx[k][1], w2 * cx[k][2]));
    float ty = fmaf(w0, cy[k][0], fmaf(w1, cy[k][1], w2 * cy[k][2]));
    float tz = fmaf(w0, cz[k][0], fmaf(w1, cz[k][1], w2 * cz[k][2]));
    r = fmaf(w, tx, r);
    g = fmaf(w, ty, g);
    b = fmaf(w, tz, b);
    denom += w;
  }

  float invd = __builtin_amdgcn_rcpf(denom);
  float bg = delta * invd;                      // background = (1,1,1)
  f4 o = {fmaf(r, invd, bg), fmaf(g, invd, bg), fmaf(b, invd, bg),
          1.0f - alpha};
  __builtin_nontemporal_store(o, (f4*)out + (size_t)p);   // streamed output
}

// ---------------------------------------------------------------------------
extern "C" void kernel_launch(void* const* d_in, const int* in_sizes, int n_in,
                              void* d_out, int out_size, void* d_ws,
                              size_t ws_size, hipStream_t stream) {
  const int*   p2f   = (const int*)d_in[0];
  const float* bary  = (const float*)d_in[1];
  const float* dists = (const float*)d_in[2];
  const float* zbuf  = (const float*)d_in[3];
  const float* vtex  = (const float*)d_in[4];
  const int*   faces = (const int*)d_in[5];
  float*       out   = (float*)d_out;
  float*       ft    = (float*)d_ws;

  const int NHW = in_sizes[0] / KF;   // N*H*W pixels
  const int F3  = in_sizes[5];        // F*3 face-vertex slots

  const int TB = 256;                 // 8 wave32 waves per block
  const int gA = (F3 + TB - 1) / TB;
  const int gB = (NHW + TB - 1) / TB;

  const bool pad4 = ws_size >= (size_t)F3 * 4u * sizeof(float);
  if (pad4) {
    build_face_tex_k<4><<<gA, TB, 0, stream>>>(vtex, faces, ft, F3);
    shade_k<4><<<gB, TB, 0, stream>>>(p2f, bary, dists, zbuf, ft, out, NHW);
  } else {
    build_face_tex_k<3><<<gA, TB, 0, stream>>>(vtex, faces, ft, F3);
    shade_k<3><<<gB, TB, 0, stream>>>(p2f, bary, dists, zbuf, ft, out, NHW);
  }
}